// STGCNLayer_65189013619314
// MI455X (gfx1250) — compile-verified
//
#include <hip/hip_runtime.h>

// ---------------- problem constants (match reference) ----------------
#define NNODES  50000
#define NBATCH  4
#define NEDGES  800000
#define CH      64          // CIN == COUT == 64
#define NTILES  3125        // 50000 / 16 exactly
#define HSTRIDE 68          // padded LDS row stride (floats) to avoid bank conflicts

typedef __attribute__((ext_vector_type(2))) float v2f;
typedef __attribute__((ext_vector_type(8))) float v8f;

static __device__ __forceinline__ v8f wmma_f32(v2f a, v2f b, v8f c) {
  // V_WMMA_F32_16X16X4_F32 : D(16x16,f32) = A(16x4,f32) * B(4x16,f32) + C
  return __builtin_amdgcn_wmma_f32_16x16x4_f32(
      /*neg_a=*/false, a, /*neg_b=*/false, b,
      /*c_mod=*/(short)0, c, /*reuse_a=*/false, /*reuse_b=*/false);
}

// ---------------- edge preprocessing ----------------
__global__ __launch_bounds__(256) void deg_kernel(const int* __restrict__ dst,
                                                  float* __restrict__ deg) {
  int e = blockIdx.x * 256 + threadIdx.x;           // E == 3125*256 exactly
  atomicAdd(&deg[dst[e]], 1.0f);
}

__global__ __launch_bounds__(256) void we_kernel(const int* __restrict__ src,
                                                 const int* __restrict__ dst,
                                                 const float* __restrict__ deg,
                                                 float* __restrict__ we) {
  int e = blockIdx.x * 256 + threadIdx.x;
  float ds = deg[src[e]];
  float dd = deg[dst[e]];
  we[e] = rsqrtf(fmaxf(ds, 1.0f)) * rsqrtf(fmaxf(dd, 1.0f));
}

// T2 = -x  (B*N*CH/4 float4 elements == 12500*256 exactly)
__global__ __launch_bounds__(256) void negcopy_kernel(const float* __restrict__ x,
                                                      float* __restrict__ t2) {
  size_t i = (size_t)blockIdx.x * 256 + threadIdx.x;
  float4 v = ((const float4*)x)[i];
  float4 r;
  r.x = -v.x; r.y = -v.y; r.z = -v.z; r.w = -v.w;
  ((float4*)t2)[i] = r;
}

// out[b, dst[e], :] += scale * w_e[e] * feat[b, src[e], :]
// grid: (3125, NBATCH*4); blockIdx.y = (b<<2)|quarter ; each thread does 16 channels
__global__ __launch_bounds__(256) void spmm_kernel(const float* __restrict__ feat,
                                                   const int* __restrict__ src,
                                                   const int* __restrict__ dst,
                                                   const float* __restrict__ we,
                                                   float scale,
                                                   float* __restrict__ out) {
  int e = blockIdx.x * 256 + threadIdx.x;
  int b = blockIdx.y >> 2;
  int q = blockIdx.y & 3;
  float w = scale * we[e];
  int s = src[e];
  int d = dst[e];
  const float4* f = (const float4*)(feat + ((size_t)b * NNODES + s) * CH + q * 16);
  float* o = out + ((size_t)b * NNODES + d) * CH + q * 16;
#pragma unroll
  for (int i = 0; i < 4; ++i) {
    float4 v = f[i];
    atomicAdd(o + 4 * i + 0, w * v.x);
    atomicAdd(o + 4 * i + 1, w * v.y);
    atomicAdd(o + 4 * i + 2, w * v.z);
    atomicAdd(o + 4 * i + 3, w * v.w);
  }
}

// ---------------- fused Chebyshev GEMM + ReLU + final Linear (WMMA f32) ----------------
// One wave per 16-node tile. Stage 1: h = relu(T0*Wc0 + T1*Wc1 + T2*Wc2 + bc) (16x64),
// staged through padded LDS; Stage 2: out = h*Wl + bl (16x64).
__global__ __launch_bounds__(256) void cheb_gemm_kernel(const float* __restrict__ T0,
                                                        const float* __restrict__ T1,
                                                        const float* __restrict__ T2,
                                                        const float* __restrict__ Wc,
                                                        const float* __restrict__ bc,
                                                        const float* __restrict__ Wl,
                                                        const float* __restrict__ bl,
                                                        float* __restrict__ out) {
  __shared__ __align__(16) float hlds[8][16 * HSTRIDE];

  const int lane = threadIdx.x & 31;
  const int wave = threadIdx.x >> 5;
  const int row  = lane & 15;       // N index within tile / M row for A loads
  const int hi   = lane >> 4;       // half-wave select

  int tile = blockIdx.x * 8 + wave;
  if (tile >= NBATCH * NTILES) return;   // wave-uniform: EXEC stays all-ones
  int b     = tile / NTILES;
  int node0 = (tile % NTILES) * 16;

  // Per-lane A row pointers (A 16x4 f32 layout: lane row, K = 2*hi + {0,1} + 4*kk)
  const size_t rowbase = ((size_t)b * NNODES + node0 + row) * CH + 2 * hi;
  const float* pA0 = T0 + rowbase;
  const float* pA1 = T1 + rowbase;
  const float* pA2 = T2 + rowbase;

  // ---- stage 1: four 16x16 output-column tiles, K=64 in 16 steps, 3 Cheb terms ----
  v8f acc[4] = {};
  for (int kk = 0; kk < 16; ++kk) {
    v2f a0 = *(const v2f*)(pA0 + 4 * kk);
    v2f a1 = *(const v2f*)(pA1 + 4 * kk);
    v2f a2 = *(const v2f*)(pA2 + 4 * kk);
    const int krow = (4 * kk + 2 * hi) * CH + row;   // B: rows K, cols N=16*j+row
#pragma unroll
    for (int j = 0; j < 4; ++j) {
      const float* w0 = Wc + 0 * CH * CH + krow + 16 * j;
      const float* w1 = Wc + 1 * CH * CH + krow + 16 * j;
      const float* w2 = Wc + 2 * CH * CH + krow + 16 * j;
      v2f b0; b0.x = w0[0]; b0.y = w0[CH];
      v2f b1; b1.x = w1[0]; b1.y = w1[CH];
      v2f b2; b2.x = w2[0]; b2.y = w2[CH];
      acc[j] = wmma_f32(a0, b0, acc[j]);
      acc[j] = wmma_f32(a1, b1, acc[j]);
      acc[j] = wmma_f32(a2, b2, acc[j]);
    }
  }

  // bias + ReLU, stash h tile in LDS (D layout: element i is [M = i + 8*hi][N = 16*j+row])
#pragma unroll
  for (int j = 0; j < 4; ++j) {
    float bcv = bc[16 * j + row];
#pragma unroll
    for (int i = 0; i < 8; ++i) {
      float h = acc[j][i] + bcv;
      h = fmaxf(h, 0.0f);
      hlds[wave][(i + 8 * hi) * HSTRIDE + 16 * j + row] = h;
    }
  }

  // ---- stage 2: out = h (16x64) * Wl (64x64) + bl ----
  v8f acc2[4] = {};
  for (int kk = 0; kk < 16; ++kk) {
    v2f a = *(const v2f*)&hlds[wave][row * HSTRIDE + 4 * kk + 2 * hi];
    const int krow = (4 * kk + 2 * hi) * CH + row;
#pragma unroll
    for (int j = 0; j < 4; ++j) {
      const float* wp = Wl + krow + 16 * j;
      v2f bf; bf.x = wp[0]; bf.y = wp[CH];
      acc2[j] = wmma_f32(a, bf, acc2[j]);
    }
  }

#pragma unroll
  for (int j = 0; j < 4; ++j) {
    float blv = bl[16 * j + row];
#pragma unroll
    for (int i = 0; i < 8; ++i) {
      out[((size_t)b * NNODES + node0 + i + 8 * hi) * CH + 16 * j + row] =
          acc2[j][i] + blv;
    }
  }
}

// ---------------- host-side launch ----------------
extern "C" void kernel_launch(void* const* d_in, const int* in_sizes, int n_in,
                              void* d_out, int out_size, void* d_ws, size_t ws_size,
                              hipStream_t stream) {
  const float* x  = (const float*)d_in[0];   // [B,N,CIN]
  const int*   src = (const int*)d_in[1];    // [E]
  const int*   dst = (const int*)d_in[2];    // [E]
  const float* Wc = (const float*)d_in[3];   // [K,CIN,COUT]
  const float* bc = (const float*)d_in[4];   // [COUT]
  const float* Wl = (const float*)d_in[5];   // [COUT,CIN]
  const float* bl = (const float*)d_in[6];   // [CIN]
  float* out = (float*)d_out;

  constexpr size_t BNC      = (size_t)NBATCH * NNODES * CH;           // 12.8M floats
  constexpr size_t OFF_DEG  = 0;
  constexpr size_t OFF_WE   = 200192;                                  // 256B aligned
  constexpr size_t OFF_T1   = OFF_WE + (size_t)NEDGES * 4;             // 3,400,192
  constexpr size_t OFF_T2   = OFF_T1 + BNC * 4;                        // +51.2MB

  char*  ws  = (char*)d_ws;
  float* deg = (float*)(ws + OFF_DEG);
  float* we  = (float*)(ws + OFF_WE);
  float* T1  = (float*)(ws + OFF_T1);
  float* T2  = (float*)(ws + OFF_T2);

  hipMemsetAsync(deg, 0, (size_t)NNODES * sizeof(float), stream);
  hipMemsetAsync(T1, 0, BNC * sizeof(float), stream);

  deg_kernel<<<NEDGES / 256, 256, 0, stream>>>(dst, deg);
  we_kernel<<<NEDGES / 256, 256, 0, stream>>>(src, dst, deg, we);
  negcopy_kernel<<<(unsigned)(BNC / 4 / 256), 256, 0, stream>>>(x, T2);

  // T1 = -A_hat @ x
  spmm_kernel<<<dim3(NEDGES / 256, NBATCH * 4), 256, 0, stream>>>(x, src, dst, we, -1.0f, T1);
  // T2 = -2 * A_hat @ T1 - x   (T2 pre-seeded with -x)
  spmm_kernel<<<dim3(NEDGES / 256, NBATCH * 4), 256, 0, stream>>>(T1, src, dst, we, -2.0f, T2);

  // fused Cheb GEMMs + ReLU + final Linear
  int tiles  = NBATCH * NTILES;                 // 12500
  int blocks = (tiles + 7) / 8;                 // 8 waves/block
  cheb_gemm_kernel<<<blocks, 256, 0, stream>>>(x, T1, T2, Wc, bc, Wl, bl, out);
}